// GraphAttentionLayer_69045894251076
// MI455X (gfx1250) — compile-verified
//
#include <hip/hip_runtime.h>
#include <hip/hip_bf16.h>

typedef float v2f __attribute__((ext_vector_type(2)));
typedef float v8f __attribute__((ext_vector_type(8)));

#define BATCH 8
#define SEQ   2048
#define FIN   256
#define FOUT  128
#define NEG_INF_F (-1e30f)

// ---------------------------------------------------------------------------
// Kernel 1: Wh = h @ W^T   (rows = B*S flattened, each block does 16 rows x 128 out)
// 8 waves per block; wave w computes the 16x16 tile at o0 = 16*w via
// V_WMMA_F32_16X16X4_F32, K-loop over FIN in steps of 4.
// ---------------------------------------------------------------------------
__global__ void wh_kernel(const float* __restrict__ h,
                          const float* __restrict__ W,
                          float* __restrict__ Wh) {
    const int t    = threadIdx.x;
    const int wv   = t >> 5;          // wave id 0..7
    const int lane = t & 31;
    const int row0 = blockIdx.x * 16; // 16 rows of flattened [B*S]
    const int o0   = wv * 16;
    const int ar   = lane & 15;       // A-matrix M / B-matrix N index
    const int koff = (lane >= 16) ? 2 : 0;

    const float* hA = h + (long)(row0 + ar) * FIN;  // row for A operand
    const float* wB = W + (long)(o0 + ar) * FIN;    // W[o][k] row for B operand

    v8f c = {0.f, 0.f, 0.f, 0.f, 0.f, 0.f, 0.f, 0.f};

    for (int k = 0; k < FIN; k += 4) {
        v2f a, b;
        // A 16x4 f32 layout: v0 -> K = k+koff, v1 -> K = k+koff+1
        a.x = hA[k + koff];
        a.y = hA[k + koff + 1];
        // B 4x16 f32 layout: lane N = ar, v0 -> K = k+koff, v1 -> K = k+koff+1
        b.x = wB[k + koff];
        b.y = wB[k + koff + 1];
        c = __builtin_amdgcn_wmma_f32_16x16x4_f32(false, a, false, b,
                                                  (short)0, c, false, false);
    }

    // C/D layout: VGPR vg, lanes 0-15 -> M=vg, lanes 16-31 -> M=vg+8; N = lane&15
#pragma unroll
    for (int vg = 0; vg < 8; vg++) {
        int m = vg + ((lane >= 16) ? 8 : 0);
        Wh[(long)(row0 + m) * FOUT + o0 + ar] = c[vg];
    }
}

// ---------------------------------------------------------------------------
// Kernel 2: e_src[i] = Wh[i,:] . a_src ; e_dst[i] = Wh[i,:] . a_dst
// ---------------------------------------------------------------------------
__global__ void esd_kernel(const float* __restrict__ Wh,
                           const float* __restrict__ a_src,
                           const float* __restrict__ a_dst,
                           float* __restrict__ e_src,
                           float* __restrict__ e_dst) {
    int i = blockIdx.x * blockDim.x + threadIdx.x; // 0 .. B*S-1
    const float* w = Wh + (long)i * FOUT;
    float s1 = 0.f, s2 = 0.f;
#pragma unroll 8
    for (int o = 0; o < FOUT; o++) {
        float v = w[o];
        s1 += v * a_src[o];
        s2 += v * a_dst[o];
    }
    e_src[i] = s1;
    e_dst[i] = s2;
}

// ---------------------------------------------------------------------------
// Kernel 3: per (b, 16-row tile): masked-LeakyReLU softmax logits, row max,
// unnormalized p written to att_out, and h' tile = (P @ Wh) via WMMA.
// Block = 256 threads = 8 waves. Row mapping for scalar phases: r = t>>4,
// 16 lanes per row striding j. WMMA phase: wave w owns o-tile w.
// ---------------------------------------------------------------------------
__global__ void gat_kernel(const float* __restrict__ Wh,
                           const float* __restrict__ e_src,
                           const float* __restrict__ e_dst,
                           const int*   __restrict__ adj,
                           const int*   __restrict__ amask,
                           float* __restrict__ att_out,
                           float* __restrict__ hprime_out,
                           float* __restrict__ s_ws) {
    const int t    = threadIdx.x;
    const int blk  = blockIdx.x;
    const int b    = blk >> 7;          // 128 row-tiles per batch
    const int i0   = (blk & 127) << 4;  // first of 16 rows in this tile
    const int r    = t >> 4;            // row within tile (0..15)
    const int jl   = t & 15;            // lane within row group
    const long base = (long)b * SEQ;

    __shared__ float eds[SEQ];       // e_dst for this batch
    __shared__ float msk[SEQ];       // attention_mask as 0/1
    __shared__ float Pb[16 * 128];   // P chunk: 16 rows x 128 j
    __shared__ float mrow[16];
    __shared__ float srow[16];

    // load e_dst + mask into LDS (coalesced)
    for (int k = 0; k < SEQ / 256; k++) {
        int j = t + 256 * k;
        eds[j] = e_dst[base + j];
        msk[j] = amask[base + j] ? 1.f : 0.f;
    }
    __syncthreads();

    const int* adjrow = adj + (base + i0 + r) * SEQ;
    const float es = e_src[base + i0 + r];

    // ---- pass 1: masked row max ----
    float m = NEG_INF_F;
    for (int k = 0; k < SEQ / 16; k++) {
        int j = jl + 16 * k;
        float e = es + eds[j];
        e = (e > 0.f) ? e : 0.2f * e;                 // LeakyReLU alpha=0.2
        bool valid = (adjrow[j] != 0) && (msk[j] != 0.f);
        e = valid ? e : NEG_INF_F;
        m = fmaxf(m, e);
    }
#pragma unroll
    for (int o = 8; o >= 1; o >>= 1) m = fmaxf(m, __shfl_xor(m, o, 32));
    if (jl == 0) mrow[r] = m;
    __syncthreads();
    m = mrow[r];

    // ---- pass 2: p = exp(e-m), write unnormalized att, WMMA P @ Wh ----
    const int wv   = t >> 5;
    const int lane = t & 31;
    const int o0   = wv * 16;
    const int ar   = lane & 15;
    const int koff = (lane >= 16) ? 2 : 0;
    const float* WhB = Wh + base * FOUT;
    float* attrow = att_out + (base + i0 + r) * SEQ;

    float ssum = 0.f;
    v8f c = {0.f, 0.f, 0.f, 0.f, 0.f, 0.f, 0.f, 0.f};

    for (int ch = 0; ch < SEQ / 128; ch++) {
        int j0 = ch * 128;
        // compute this P chunk (16 rows x 128 j)
#pragma unroll
        for (int k = 0; k < 8; k++) {
            int jj = jl + 16 * k;
            int j  = j0 + jj;
            float e = es + eds[j];
            e = (e > 0.f) ? e : 0.2f * e;
            bool valid = (adjrow[j] != 0) && (msk[j] != 0.f);
            float p = valid ? __expf(e - m) : 0.f;
            ssum += p;
            Pb[r * 128 + jj] = p;
            attrow[j] = p;   // unnormalized; kernel 4 divides by row sum
        }
        __syncthreads();

        // WMMA: accumulate c += P[16 x 128chunk] @ Wh[128chunk x 16]
        const float* Pa = Pb + ar * 128;
        for (int k = 0; k < 128; k += 4) {
            v2f a, bb;
            a.x = Pa[k + koff];
            a.y = Pa[k + koff + 1];
            const float* wp = WhB + (long)(j0 + k + koff) * FOUT + o0 + ar;
            bb.x = wp[0];
            bb.y = wp[FOUT];
            c = __builtin_amdgcn_wmma_f32_16x16x4_f32(false, a, false, bb,
                                                      (short)0, c, false, false);
        }
        __syncthreads();
    }

    // reduce row sums across the 16-lane row group
#pragma unroll
    for (int o = 8; o >= 1; o >>= 1) ssum += __shfl_xor(ssum, o, 32);
    if (jl == 0) {
        srow[r] = ssum;
        s_ws[base + i0 + r] = ssum;
    }
    __syncthreads();

    // h' = elu(c / s); C/D layout: VGPR vg -> M = vg (+8 for lanes 16-31), N = ar
#pragma unroll
    for (int vg = 0; vg < 8; vg++) {
        int mi = vg + ((lane >= 16) ? 8 : 0);
        float s = srow[mi];
        float x = (s > 0.f) ? (c[vg] / s) : 0.f;
        x = (x > 0.f) ? x : (__expf(x) - 1.f);   // ELU
        hprime_out[(base + i0 + mi) * FOUT + o0 + ar] = x;
    }
}

// ---------------------------------------------------------------------------
// Kernel 4: normalize attention rows: att /= s (s<=0 -> 0)
// ---------------------------------------------------------------------------
__global__ void norm_kernel(float* __restrict__ att,
                            const float* __restrict__ s_ws,
                            long n) {
    long stride = (long)gridDim.x * blockDim.x;
    for (long idx = (long)blockIdx.x * blockDim.x + threadIdx.x; idx < n; idx += stride) {
        long row = idx >> 11;          // / SEQ
        float s = s_ws[row];
        float p = att[idx];
        att[idx] = (s > 0.f) ? (p / s) : 0.f;
    }
}

// ---------------------------------------------------------------------------
extern "C" void kernel_launch(void* const* d_in, const int* in_sizes, int n_in,
                              void* d_out, int out_size, void* d_ws, size_t ws_size,
                              hipStream_t stream) {
    const float* h     = (const float*)d_in[0];   // [B,S,Fin]
    const int*   adj   = (const int*)  d_in[1];   // [B,S,S]
    const int*   amask = (const int*)  d_in[2];   // [B,S]
    const float* W     = (const float*)d_in[3];   // [Fout,Fin]
    const float* a_src = (const float*)d_in[4];   // [Fout,1]
    const float* a_dst = (const float*)d_in[5];   // [Fout,1]

    const long nRows = (long)BATCH * SEQ;                 // 16384
    float* hprime = (float*)d_out;                        // [B,S,Fout]
    float* att    = (float*)d_out + nRows * FOUT;         // [B,S,S]

    // workspace layout
    float* Wh    = (float*)d_ws;                          // nRows*FOUT
    float* e_src = Wh + nRows * FOUT;                     // nRows
    float* e_dst = e_src + nRows;                         // nRows
    float* s_ws  = e_dst + nRows;                         // nRows

    // 1) Wh = h @ W^T     (1024 blocks x 256 threads; 16 rows, 8 o-tiles each)
    wh_kernel<<<dim3((unsigned)(nRows / 16)), dim3(256), 0, stream>>>(h, W, Wh);

    // 2) e_src / e_dst
    esd_kernel<<<dim3((unsigned)(nRows / 256)), dim3(256), 0, stream>>>(
        Wh, a_src, a_dst, e_src, e_dst);

    // 3) softmax logits + P@Wh  (1024 blocks: 8 batches x 128 row-tiles)
    gat_kernel<<<dim3((unsigned)(nRows / 16)), dim3(256), 0, stream>>>(
        Wh, e_src, e_dst, adj, amask, att, hprime, s_ws);

    // 4) normalize attention
    long nAtt = (long)BATCH * SEQ * SEQ;
    norm_kernel<<<dim3(2048), dim3(256), 0, stream>>>(att, s_ws, nAtt);
}